// S4Layer_27977416966200
// MI455X (gfx1250) — compile-verified
//
#include <hip/hip_runtime.h>
#include <math.h>

#define Hc   128
#define Lsz  8192
#define Nst  64

typedef __attribute__((ext_vector_type(2))) float v2f;
typedef __attribute__((ext_vector_type(8))) float v8f;

// fast hardware reciprocal (v_rcp_f32, ~1 ulp) — avoids the IEEE div expansion
__device__ __forceinline__ float frcp(float x) { return __builtin_amdgcn_rcpf(x); }

// ---------------------------------------------------------------------------
// Kernel 1: per-channel prep.  B = Vc @ B_raw, then pack the 4 complex cauchy
// weight vectors into a (128 x 16) real matrix Bm per channel:
//   row 2n   : [ w00r  w00i  w01r  w01i  w10r  w10i  w11r  w11i  0...0 ]
//   row 2n+1 : [-w00i  w00r -w01i  w01r -w10i  w10r -w11i  w11r  0...0 ]
// so that  K(16xL-tile) = A(recip re/im) x Bm  gives re/im of k00,k01,k10,k11.
// ---------------------------------------------------------------------------
__global__ void s4_prep(const float* __restrict__ Ct_ri,
                        const float* __restrict__ B_raw,
                        const float* __restrict__ log_step,
                        const float* __restrict__ p_re, const float* __restrict__ p_im,
                        const float* __restrict__ q_re, const float* __restrict__ q_im,
                        const float* __restrict__ Vc_re, const float* __restrict__ Vc_im,
                        float* __restrict__ stepv, float* __restrict__ Bm)
{
    const int h = blockIdx.x;
    const int n = threadIdx.x;

    float br = 0.f, bi = 0.f;
    for (int m = 0; m < Nst; ++m) {
        float v = B_raw[h * Nst + m];
        br += Vc_re[n * Nst + m] * v;
        bi += Vc_im[n * Nst + m] * v;
    }
    const float cr = Ct_ri[(h * Nst + n) * 2 + 0];
    const float ci = Ct_ri[(h * Nst + n) * 2 + 1];
    const float pr = p_re[n], pi = p_im[n];
    const float qr = q_re[n], qi = q_im[n];

    // conj(Ct)*B, conj(Ct)*p, conj(q)*B, conj(q)*p
    const float w00r = cr*br + ci*bi, w00i = cr*bi - ci*br;
    const float w01r = cr*pr + ci*pi, w01i = cr*pi - ci*pr;
    const float w10r = qr*br + qi*bi, w10i = qr*bi - qi*br;
    const float w11r = qr*pr + qi*pi, w11i = qr*pi - qi*pr;

    float* re = Bm + ((size_t)h * 128 + 2 * n) * 16;
    float* ro = re + 16;
    re[0]= w00r; re[1]= w00i; re[2]= w01r; re[3]= w01i;
    re[4]= w10r; re[5]= w10i; re[6]= w11r; re[7]= w11i;
    ro[0]=-w00i; ro[1]= w00r; ro[2]=-w01i; ro[3]= w01r;
    ro[4]=-w10i; ro[5]= w10r; ro[6]=-w11i; ro[7]= w11r;
#pragma unroll
    for (int c = 8; c < 16; ++c) { re[c] = 0.f; ro[c] = 0.f; }

    if (n == 0) stepv[h] = expf(log_step[h]);
}

// ---------------------------------------------------------------------------
// Kernel 2: cauchy dot via fp32 WMMA.  One wave per (h, 16-row l tile).
// A fragment (16x4 f32): lanes 0-15 hold K=0/K=1 of the k-chunk, lanes 16-31
// hold K=2/K=3 (ISA 7.12.2).  Chunk 4t covers n=2t (low lanes) and n=2t+1
// (high lanes): each lane computes recip = 1/(g - Lambda_n) and feeds (rr,ri).
// B fragment mirrors the same lane->K mapping, loaded from LDS-cached Bm.
// ---------------------------------------------------------------------------
__global__ void __launch_bounds__(32) s4_cauchy(
        const float* __restrict__ Lre, const float* __restrict__ Lim,
        const float* __restrict__ stepv, const float* __restrict__ Bm,
        float2* __restrict__ at)
{
    __shared__ float Bsh[128 * 16];
    __shared__ float Csh[16 * 16];

    const int h    = blockIdx.x;
    const int l0   = blockIdx.y * 16;
    const int lane = threadIdx.x;
    const int ml   = lane & 15;
    const bool hi  = lane >= 16;

    for (int i = lane; i < 128 * 16; i += 32)
        Bsh[i] = Bm[(size_t)h * 2048 + i];

    // per-lane g(h,l) and c(l) exactly as the reference computes them
    const int   myl   = l0 + ml;
    const float theta = (float)myl * (6.283185307179586f / 8192.0f);
    const float om_r  = cosf(theta);
    const float om_i  = -sinf(theta);            // omega = e^{-i theta}
    const float nr = 1.f - om_r, ni = -om_i;     // 1 - omega
    const float dr_ = 1.f + om_r, di_ = om_i;    // 1 + omega
    const float invden = frcp(dr_*dr_ + di_*di_);
    const float rat_r = (nr*dr_ + ni*di_) * invden;
    const float rat_i = (ni*dr_ - nr*di_) * invden;
    const float tos = 2.0f * frcp(stepv[h]);
    const float g_r = tos * rat_r;
    const float g_i = tos * rat_i;
    const float c_r =  2.0f * dr_ * invden;      // c = 2/(1+omega)
    const float c_i = -2.0f * di_ * invden;

    __syncthreads();

    v8f acc = {0.f, 0.f, 0.f, 0.f, 0.f, 0.f, 0.f, 0.f};
    const int roff = hi ? 2 : 0;
#pragma unroll 8
    for (int t = 0; t < 32; ++t) {
        const int n = 2 * t + (hi ? 1 : 0);
        const float dr  = g_r - Lre[n];
        const float di  = g_i - Lim[n];
        const float inv = frcp(dr*dr + di*di);   // v_rcp_f32: co-issues w/ VALU
        v2f a; a.x = dr * inv; a.y = -di * inv;  // rr, ri
        v2f b;
        b.x = Bsh[(4*t + roff    ) * 16 + ml];
        b.y = Bsh[(4*t + roff + 1) * 16 + ml];
        acc = __builtin_amdgcn_wmma_f32_16x16x4_f32(
                  false, a, false, b, (short)0, acc, false, false);
    }

    // C tile: VGPR r, lane -> (M = r + 8*hi, N = ml)
#pragma unroll
    for (int r = 0; r < 8; ++r)
        Csh[(r + (hi ? 8 : 0)) * 16 + ml] = acc[r];
    __syncthreads();

    if (lane < 16) {
        const float* row = &Csh[ml * 16];
        const float k00r=row[0], k00i=row[1], k01r=row[2], k01i=row[3];
        const float k10r=row[4], k10i=row[5], k11r=row[6], k11i=row[7];
        const float kdr = 1.f + k11r, kdi = k11i;
        const float invd = frcp(kdr*kdr + kdi*kdi);
        const float qr_ = k01r*k10r - k01i*k10i;
        const float qi_ = k01r*k10i + k01i*k10r;
        const float tr = (qr_*kdr + qi_*kdi) * invd;
        const float ti = (qi_*kdr - qr_*kdi) * invd;
        const float sr = k00r - tr, si = k00i - ti;
        at[(size_t)h * Lsz + myl] = make_float2(c_r*sr - c_i*si,
                                                c_r*si + c_i*sr);
    }
}

// ---------------------------------------------------------------------------
// Kernel 3: spectral part, entirely in LDS (128 KB dynamic, CDNA5 WGP = 320KB).
// ---------------------------------------------------------------------------
__device__ inline void block_fft(float2* s, int n, int bits, float sign,
                                 int tid, int nth)
{
    for (int i = tid; i < n; i += nth) {
        const int r = (int)(__brev((unsigned)i) >> (32 - bits));
        if (r > i) { float2 t = s[i]; s[i] = s[r]; s[r] = t; }
    }
    __syncthreads();
    for (int stg = 1; stg <= bits; ++stg) {
        const int m = 1 << stg;
        const int half = m >> 1;
        const float base = sign * 6.283185307179586f / (float)m;
        for (int j = tid; j < (n >> 1); j += nth) {
            const int grp = j >> (stg - 1);
            const int pos = j & (half - 1);
            const int i0 = grp * m + pos;
            const int i1 = i0 + half;
            const float ang = base * (float)pos;
            const float wr = cosf(ang), wi = sinf(ang);
            const float2 a = s[i0], b = s[i1];
            const float tr = wr*b.x - wi*b.y;
            const float ti = wr*b.y + wi*b.x;
            s[i1] = make_float2(a.x - tr, a.y - ti);
            s[i0] = make_float2(a.x + tr, a.y + ti);
        }
        __syncthreads();
    }
}

__global__ void __launch_bounds__(256) s4_fft(const float2* __restrict__ at,
                                              const float* __restrict__ u,
                                              const float* __restrict__ D,
                                              float* __restrict__ y)
{
    extern __shared__ float2 smem[];
    const int h   = blockIdx.x;
    const int tid = threadIdx.x;
    const int nth = 256;

    for (int i = tid; i < Lsz; i += nth)
        smem[i] = at[(size_t)h * Lsz + i];
    // warm L2/WGP$ for the u re-reads that follow the long IFFT phase
    for (int i = tid; i < Lsz; i += nth)
        __builtin_prefetch(&u[(size_t)h * Lsz + i], 0, 3);
    __syncthreads();

    // K = Re(IFFT_8192(at_roots)) / 8192
    block_fft(smem, Lsz, 13, +1.0f, tid, nth);
    float kreg[32];
#pragma unroll
    for (int ii = 0; ii < 32; ++ii)
        kreg[ii] = smem[tid + ii * 256].x * (1.0f / 8192.0f);
    __syncthreads();

    // z = u + i*K, zero padded to 2L   (one complex FFT = both real FFTs)
#pragma unroll
    for (int ii = 0; ii < 32; ++ii) {
        const int idx = tid + ii * 256;
        smem[idx]       = make_float2(u[(size_t)h * Lsz + idx], kreg[ii]);
        smem[idx + Lsz] = make_float2(0.f, 0.f);
    }
    __syncthreads();

    block_fft(smem, 2 * Lsz, 14, -1.0f, tid, nth);

    // Hermitian split + product: U = (Z+conj(Zr))/2, Kh = (Z-conj(Zr))/(2i)
    for (int k = tid; k <= Lsz; k += nth) {
        const int kp = (2 * Lsz - k) & (2 * Lsz - 1);
        const float2 z1 = smem[k];
        const float2 z2 = smem[kp];
        const float bx = z2.x, by = -z2.y;
        const float ux = 0.5f * (z1.x + bx), uy = 0.5f * (z1.y + by);
        const float dx = z1.x - bx,          dy = z1.y - by;
        const float vx = 0.5f * dy,          vy = -0.5f * dx;  // -0.5i*(dx+idy)
        const float px = ux*vx - uy*vy;
        const float py = ux*vy + uy*vx;
        smem[k] = make_float2(px, py);
        if (kp != k) smem[kp] = make_float2(px, -py);
    }
    __syncthreads();

    block_fft(smem, 2 * Lsz, 14, +1.0f, tid, nth);

    const float dsc = D[h];
    for (int i = tid; i < Lsz; i += nth) {
        const float uv = u[(size_t)h * Lsz + i];
        y[(size_t)h * Lsz + i] = smem[i].x * (1.0f / 16384.0f) + dsc * uv;
    }
}

// ---------------------------------------------------------------------------
extern "C" void kernel_launch(void* const* d_in, const int* in_sizes, int n_in,
                              void* d_out, int out_size, void* d_ws, size_t ws_size,
                              hipStream_t stream)
{
    const float* u        = (const float*)d_in[0];
    const float* Ct_ri    = (const float*)d_in[1];
    const float* B_raw    = (const float*)d_in[2];
    const float* D        = (const float*)d_in[3];
    const float* log_step = (const float*)d_in[4];
    const float* Lre      = (const float*)d_in[5];
    const float* Lim      = (const float*)d_in[6];
    const float* p_re     = (const float*)d_in[7];
    const float* p_im     = (const float*)d_in[8];
    const float* q_re     = (const float*)d_in[9];
    const float* q_im     = (const float*)d_in[10];
    const float* Vc_re    = (const float*)d_in[11];
    const float* Vc_im    = (const float*)d_in[12];

    float*  ws    = (float*)d_ws;
    float*  stepv = ws;                                  // 128 f32
    float*  Bm    = ws + 128;                            // 128*128*16 f32 (1 MB)
    float2* at    = (float2*)(ws + 128 + 128 * 2048);    // 128*8192 cf32 (8 MB)

    s4_prep<<<dim3(Hc), dim3(Nst), 0, stream>>>(
        Ct_ri, B_raw, log_step, p_re, p_im, q_re, q_im, Vc_re, Vc_im,
        stepv, Bm);

    s4_cauchy<<<dim3(Hc, Lsz / 16), dim3(32), 0, stream>>>(
        Lre, Lim, stepv, Bm, at);

    s4_fft<<<dim3(Hc), dim3(256), 2 * Lsz * sizeof(float2), stream>>>(
        at, u, D, (float*)d_out);
}